// PershomReadout_71554155151373
// MI455X (gfx1250) — compile-verified
//
#include <hip/hip_runtime.h>
#include <hip/hip_bf16.h>

typedef __attribute__((ext_vector_type(2))) float v2f;
typedef __attribute__((ext_vector_type(8))) float v8f;

#define B_      32
#define K_      64
#define N0_     20000
#define N1_     10000
#define CHUNK_  2048
#define CH_A    10          // chunks covering 20000 raw points
#define CH_E    5           // chunks covering each 10000-point ext segment
#define CH_TOT  20          // per (b, phase)

static __device__ __forceinline__ int imin(int a, int b) { return a < b ? a : b; }

// val = 1/(1+|px-cx|+|py-cy|) - 1/(1+|r - (|px-cx|+|py-cy|)|)
static __device__ __forceinline__ float rhat(float px, float py, float cx, float cy, float r) {
    float nrm = __builtin_fabsf(px - cx) + __builtin_fabsf(py - cy);
    return __builtin_amdgcn_rcpf(1.0f + nrm)
         - __builtin_amdgcn_rcpf(1.0f + __builtin_fabsf(r - nrm));
}

// Branch-free inner loop, specialized per coordinate-transform mode.
// MODE 0: point = (x, y);  MODE 1: (x, 1-x);  MODE 2: (y, 1-y).
template <int MODE>
static __device__ __forceinline__ void point_loop(
    const float* __restrict__ segPtr, int wstart, int wend, int laneOff,
    float2 c0, float2 c1, float2 c2, float2 c3, float r,
    v8f& acc0, v8f& acc1, v8f& acc2, v8f& acc3) {

    const v2f onesB = {1.0f, 1.0f};              // B matrix = all ones (4x16)
    const float4* __restrict__ p4 =
        (const float4*)(segPtr + 2 * (wstart + laneOff));
    const int iters = (wend - wstart) >> 2;      // group of 4 points / iter

    for (int i = 0; i < iters; ++i) {
        const float4 v = *p4;
        p4 += 2;                                 // 4 points = 32 bytes / iter

        float ax, ay, bx, by;
        if (MODE == 0)      { ax = v.x; ay = v.y;        bx = v.z; by = v.w;        }
        else if (MODE == 1) { ax = v.x; ay = 1.0f - v.x; bx = v.z; by = 1.0f - v.z; }
        else                { ax = v.y; ay = 1.0f - v.y; bx = v.w; by = 1.0f - v.w; }

        v2f A;
        A.x = rhat(ax, ay, c0.x, c0.y, r);
        A.y = rhat(bx, by, c0.x, c0.y, r);
        acc0 = __builtin_amdgcn_wmma_f32_16x16x4_f32(false, A, false, onesB, (short)0, acc0, false, false);

        A.x = rhat(ax, ay, c1.x, c1.y, r);
        A.y = rhat(bx, by, c1.x, c1.y, r);
        acc1 = __builtin_amdgcn_wmma_f32_16x16x4_f32(false, A, false, onesB, (short)0, acc1, false, false);

        A.x = rhat(ax, ay, c2.x, c2.y, r);
        A.y = rhat(bx, by, c2.x, c2.y, r);
        acc2 = __builtin_amdgcn_wmma_f32_16x16x4_f32(false, A, false, onesB, (short)0, acc2, false, false);

        A.x = rhat(ax, ay, c3.x, c3.y, r);
        A.y = rhat(bx, by, c3.x, c3.y, r);
        acc3 = __builtin_amdgcn_wmma_f32_16x16x4_f32(false, A, false, onesB, (short)0, acc3, false, false);
    }
}

// Stage 1: each block = (b, phase, chunk). 8 waves x 256 points each.
// WMMA f32 16x16x4 with B=ones performs the 4-point reduction for 16 centers
// per instruction; 4 accumulators cover K=64 centers.
__global__ __launch_bounds__(256) void rh_main(
    const float* __restrict__ b0up, const float* __restrict__ b0dn,
    const float* __restrict__ e0,   const float* __restrict__ e1,
    const float* __restrict__ centers, const float* __restrict__ radius,
    float* __restrict__ partial /* [B][2][CH_TOT][K] */) {

    __shared__ float lds[8][K_];

    const int task  = blockIdx.x;
    const int chunk = task % CH_TOT;
    const int phase = (task / CH_TOT) & 1;      // 0 = up, 1 = down
    const int b     = task / (CH_TOT * 2);
    const int lane  = threadIdx.x & 31;
    const int wave  = threadIdx.x >> 5;

    // Segment select (all boundaries are multiples of 4 -> uniform groups).
    const float* segPtr;
    int segLen, mode, segStart;
    if (chunk < CH_A) {
        segPtr   = (phase == 0 ? b0up : b0dn) + (size_t)b * N0_ * 2;
        segLen   = N0_;
        mode     = 0;                            // point = (x, y)
        segStart = chunk * CHUNK_;
    } else if (chunk < CH_A + CH_E) {
        segPtr   = e0 + (size_t)b * N1_ * 2;
        segLen   = N1_;
        mode     = (phase == 0) ? 2 : 1;         // up:(y,1-y)  down:(x,1-x)
        segStart = (chunk - CH_A) * CHUNK_;
    } else {
        segPtr   = e1 + (size_t)b * N1_ * 2;
        segLen   = N1_;
        mode     = (phase == 0) ? 2 : 1;
        segStart = (chunk - CH_A - CH_E) * CHUNK_;
    }
    const int segEnd = imin(segStart + CHUNK_, segLen);

    const float r = __builtin_fabsf(radius[0]);

    // Per-lane centers: row M = lane%16 for center group g -> center g*16 + M.
    const int m = lane & 15;
    const float2 c0 = ((const float2*)centers)[0 * 16 + m];
    const float2 c1 = ((const float2*)centers)[1 * 16 + m];
    const float2 c2 = ((const float2*)centers)[2 * 16 + m];
    const float2 c3 = ((const float2*)centers)[3 * 16 + m];

    v8f acc0 = {}, acc1 = {}, acc2 = {}, acc3 = {};

    // A layout (16x4 f32): lanes 0-15 hold K=0 (vgpr0), K=1 (vgpr1);
    // lanes 16-31 hold K=2, K=3. So this lane evaluates points p+laneOff, +1.
    const int laneOff = (lane >> 4) << 1;        // 0 or 2

    const int wstart = segStart + wave * 256;
    const int wend   = imin(wstart + 256, segEnd);

    if (mode == 0)
        point_loop<0>(segPtr, wstart, wend, laneOff, c0, c1, c2, c3, r, acc0, acc1, acc2, acc3);
    else if (mode == 1)
        point_loop<1>(segPtr, wstart, wend, laneOff, c0, c1, c2, c3, r, acc0, acc1, acc2, acc3);
    else
        point_loop<2>(segPtr, wstart, wend, laneOff, c0, c1, c2, c3, r, acc0, acc1, acc2, acc3);

    // D layout: vgpr rI, lanes 0-15 -> row rI, lanes 16-31 -> row rI+8;
    // all 16 columns hold the identical row sum, so lanes 0 and 16 suffice.
    if (lane == 0 || lane == 16) {
        const int kb = (lane == 16) ? 8 : 0;
#pragma unroll
        for (int rI = 0; rI < 8; ++rI) {
            lds[wave][ 0 + kb + rI] = acc0[rI];
            lds[wave][16 + kb + rI] = acc1[rI];
            lds[wave][32 + kb + rI] = acc2[rI];
            lds[wave][48 + kb + rI] = acc3[rI];
        }
    }
    __syncthreads();

    if (threadIdx.x < K_) {
        const int k = threadIdx.x;
        float s = 0.0f;
#pragma unroll
        for (int w = 0; w < 8; ++w) s += lds[w][k];   // fixed order: deterministic
        partial[((size_t)((b * 2 + phase) * CH_TOT + chunk)) * K_ + k] = s;
    }
}

// Stage 2: per-batch chunk reduction (fixed order), writes x and per-batch tpl partial.
__global__ __launch_bounds__(64) void rh_stage2(
    const float* __restrict__ partial, float* __restrict__ x_out,
    float* __restrict__ tplpart) {
    __shared__ float lds[K_];
    const int b = blockIdx.x;
    const int k = threadIdx.x;

    const float* pu = partial + ((size_t)(b * 2 + 0) * CH_TOT) * K_ + k;
    const float* pd = partial + ((size_t)(b * 2 + 1) * CH_TOT) * K_ + k;
    float u = 0.0f, d = 0.0f;
#pragma unroll
    for (int c = 0; c < CH_TOT; ++c) { u += pu[c * K_]; d += pd[c * K_]; }

    x_out[b * (2 * K_) + k]      = u;   // up_out
    x_out[b * (2 * K_) + K_ + k] = d;   // down_out

    const float df = u - d;
    lds[k] = df * df;
    __syncthreads();
    if (k == 0) {
        float s = 0.0f;
#pragma unroll
        for (int i = 0; i < K_; ++i) s += lds[i];
        tplpart[b] = s;
    }
}

// Stage 3: final tpl scalar.
__global__ __launch_bounds__(32) void rh_stage3(
    const float* __restrict__ tplpart, float* __restrict__ out_tpl) {
    if (threadIdx.x == 0) {
        float s = 0.0f;
#pragma unroll
        for (int b = 0; b < B_; ++b) s += tplpart[b];
        out_tpl[0] = -s;
    }
}

extern "C" void kernel_launch(void* const* d_in, const int* in_sizes, int n_in,
                              void* d_out, int out_size, void* d_ws, size_t ws_size,
                              hipStream_t stream) {
    const float* b0up    = (const float*)d_in[0];
    const float* b0dn    = (const float*)d_in[1];
    const float* e0      = (const float*)d_in[2];
    const float* e1      = (const float*)d_in[3];
    const float* centers = (const float*)d_in[4];
    const float* radius  = (const float*)d_in[5];

    float* xout    = (float*)d_out;                                  // [32][128]
    float* partial = (float*)d_ws;                                   // [32][2][20][64]
    float* tplpart = (float*)((char*)d_ws +
                      (size_t)B_ * 2 * CH_TOT * K_ * sizeof(float)); // [32]

    rh_main<<<dim3(B_ * 2 * CH_TOT), 256, 0, stream>>>(
        b0up, b0dn, e0, e1, centers, radius, partial);
    rh_stage2<<<dim3(B_), 64, 0, stream>>>(partial, xout, tplpart);
    rh_stage3<<<dim3(1), 32, 0, stream>>>(tplpart, xout + (size_t)B_ * 2 * K_);
}